// GraphSAGELayer_19155554140771
// MI455X (gfx1250) — compile-verified
//
#include <hip/hip_runtime.h>

typedef __attribute__((ext_vector_type(2))) float v2f;
typedef __attribute__((ext_vector_type(8))) float v8f;

#define NODES_PER_BLOCK 16
#define DEGREE 16
#define DIM 256
#define NORM_EPS 1e-12f

__global__ __launch_bounds__(256) void sage_fused_kernel(
    const int*   __restrict__ neighbors,  // [n, 16]
    const float* __restrict__ emb,        // [n, 256]
    const float* __restrict__ W,          // [256, 256] row-major [out, in]
    float*       __restrict__ out,        // [n, 256]
    int n_nodes)
{
    __shared__ float lds_means[NODES_PER_BLOCK * DIM];   // 16 KB
    __shared__ float lds_out  [NODES_PER_BLOCK * DIM];   // 16 KB
    __shared__ int   lds_nbr  [NODES_PER_BLOCK * DEGREE];
    __shared__ float lds_norm [NODES_PER_BLOCK];
    __shared__ float lds_scale[NODES_PER_BLOCK];

    const int tid       = threadIdx.x;
    const int node_base = blockIdx.x * NODES_PER_BLOCK;

    if (tid < NODES_PER_BLOCK) lds_norm[tid] = 0.0f;

    // ---- stage neighbor indices (256 ints == blockDim) ----
    {
        const int node = node_base + (tid >> 4);
        int idx = 0;
        if (node < n_nodes) idx = neighbors[(size_t)node_base * DEGREE + tid];
        if (idx < 0) idx = 0;
        if (idx >= n_nodes) idx = n_nodes - 1;
        lds_nbr[tid] = idx;
    }
    __syncthreads();

    // ---- gather + mean into LDS (float4 / global_load_b128 path) ----
    {
        const int grp = tid >> 6;          // 4 groups of 64 lanes, 4 nodes each
        const int c4  = (tid & 63) << 2;   // 4 consecutive columns per thread
        #pragma unroll
        for (int i = 0; i < 4; ++i) {
            const int node_l = grp * 4 + i;
            float4 acc = make_float4(0.f, 0.f, 0.f, 0.f);
            #pragma unroll
            for (int j = 0; j < DEGREE; ++j) {
                const int row = lds_nbr[node_l * DEGREE + j];
                const float4 v =
                    *reinterpret_cast<const float4*>(emb + (size_t)row * DIM + c4);
                acc.x += v.x; acc.y += v.y; acc.z += v.z; acc.w += v.w;
            }
            const float s = 1.0f / (float)DEGREE;
            *reinterpret_cast<float4*>(&lds_means[node_l * DIM + c4]) =
                make_float4(acc.x * s, acc.y * s, acc.z * s, acc.w * s);
        }
    }
    __syncthreads();

    // ---- GEMM via V_WMMA_F32_16X16X4_F32: 8 waves x 2 N-tiles ----
    const int wave  = tid >> 5;
    const int lane  = tid & 31;
    const int m     = lane & 15;            // A row (node)
    const int khalf = (lane >> 4) << 1;     // K sub-offset: 0 or 2

    for (int t = 0; t < 2; ++t) {
        const int nt     = wave * 2 + t;            // 0..15 output tile
        const int n_glob = nt * 16 + (lane & 15);   // output feature (W row)
        v8f c = {};
        #pragma unroll 8
        for (int k = 0; k < DIM; k += 4) {
            v2f a = *reinterpret_cast<const v2f*>(&lds_means[m * DIM + k + khalf]);
            v2f b = *reinterpret_cast<const v2f*>(W + (size_t)n_glob * DIM + k + khalf);
            c = __builtin_amdgcn_wmma_f32_16x16x4_f32(
                    false, a, false, b, (short)0, c, false, false);
        }
        // ReLU + scatter tile into LDS (C/D layout: VGPR i -> row i + 8*(lane>=16))
        #pragma unroll
        for (int i = 0; i < 8; ++i) {
            float v = c[i] > 0.f ? c[i] : 0.f;
            const int row = i + ((lane >> 4) << 3);
            const int col = nt * 16 + (lane & 15);
            lds_out[row * DIM + col] = v;
        }
    }
    __syncthreads();

    // ---- row sum-of-squares: 16 threads per row, ds_add_f32 reduction ----
    {
        const int r  = tid >> 4;
        const int cb = (tid & 15) << 4;
        float p = 0.f;
        #pragma unroll
        for (int i = 0; i < 16; ++i) {
            const float v = lds_out[r * DIM + cb + i];
            p += v * v;
        }
        atomicAdd(&lds_norm[r], p);
    }
    __syncthreads();

    if (tid < NODES_PER_BLOCK) {
        const float nrm = sqrtf(lds_norm[tid]);
        lds_scale[tid] = 1.0f / fmaxf(nrm, NORM_EPS);
    }
    __syncthreads();

    // ---- normalized, fully coalesced store ----
    #pragma unroll
    for (int r = 0; r < NODES_PER_BLOCK; ++r) {
        const int node = node_base + r;
        if (node < n_nodes) {
            out[(size_t)node * DIM + tid] = lds_out[r * DIM + tid] * lds_scale[r];
        }
    }
}

extern "C" void kernel_launch(void* const* d_in, const int* in_sizes, int n_in,
                              void* d_out, int out_size, void* d_ws, size_t ws_size,
                              hipStream_t stream) {
    (void)n_in; (void)out_size; (void)d_ws; (void)ws_size;
    const int*   neighbors = (const int*)d_in[0];
    const float* emb       = (const float*)d_in[1];
    const float* W         = (const float*)d_in[2];
    float*       out       = (float*)d_out;

    const int n_nodes = in_sizes[0] / DEGREE;          // neighbors is [n, 16]
    const int blocks  = (n_nodes + NODES_PER_BLOCK - 1) / NODES_PER_BLOCK;

    sage_fused_kernel<<<blocks, 256, 0, stream>>>(neighbors, emb, W, out, n_nodes);
}